// REN_IQC_gamma_53343493816808
// MI455X (gfx1250) — compile-verified
//
#include <hip/hip_runtime.h>
#include <hip/hip_bf16.h>
#include <math.h>

// ---------------------------------------------------------------------------
// REN scan for MI455X (gfx1250, wave32, WMMA).
// Sizes: BATCH=512, HORIZON=512, DIM_IN=64, DIM_OUT=64, DIM_X=256, DIM_NL=128
//   * setup kernels: H = X^T X + eps I, extract blocks, Gauss-Jordan invert E,
//     fold E_inv into weights (Axt = (E_inv Fm)^T etc., stored k-major so WMMA
//     B-fragments load contiguously).
//   * ONE persistent scan kernel: 32 workgroups, each owns a 16-row batch slab
//     and runs all 511 steps with x/w/base resident in LDS. All GEMM work uses
//     v_wmma_f32_16x16x4_f32. Time loop unrolled 2x so the x ping-pong buffers
//     are compile-time LDS addresses (no pointer selects, direct ds_ ops).
// ---------------------------------------------------------------------------

typedef __attribute__((ext_vector_type(2))) float v2f;
typedef __attribute__((ext_vector_type(8))) float v8f;

#define LDX 260   // padded LDS stride for 16x256 x-slab (bank-conflict free)
#define LDW 136   // padded LDS stride for 16x128 w/base slabs

// NT adjacent 16x16 output tiles sharing one A-fragment stream:
//   D[i] += A(16xK) * Bt(Kx16 at column offset 16*i)
// A:  row-major activations, A[mrow*lda + k]
// Bt: k-major weights,       Bt[k*ldb + n], caller pre-offsets base column.
// Lane mapping per CDNA5 ISA 7.12.2:
//   A frag : lane l holds A[l&15, k0+2*(l>>4)] and A[l&15, k0+2*(l>>4)+1]
//   B frag : lane l holds Bt[k0+2*(l>>4), l&15] and Bt[k0+2*(l>>4)+1, l&15]
//   C/D    : VGPR r = element (M = r + 8*(l>>4), N = l&15)
template <int NT>
__device__ __forceinline__ void wmma_multi(const float* __restrict__ A, int lda,
                                           const float* __restrict__ Bt, int ldb,
                                           int K, int mrow, int hf, v8f* acc) {
  for (int k0 = 0; k0 < K; k0 += 4) {
    int ka = k0 + 2 * hf;
    v2f a;
    a.x = A[mrow * lda + ka];
    a.y = A[mrow * lda + ka + 1];
#pragma unroll
    for (int i = 0; i < NT; ++i) {
      v2f b;
      b.x = Bt[ka * ldb + i * 16 + mrow];
      b.y = Bt[(ka + 1) * ldb + i * 16 + mrow];
      acc[i] = __builtin_amdgcn_wmma_f32_16x16x4_f32(false, a, false, b,
                                                     (short)0, acc[i],
                                                     false, false);
    }
  }
}

// ---------------------------------------------------------------------------
// Setup kernels
// ---------------------------------------------------------------------------

// H = X^T X + eps*I   (640x640)
__global__ void k_gram(const float* __restrict__ X, float* __restrict__ H) {
  int i = blockIdx.y * 16 + threadIdx.y;
  int j = blockIdx.x * 16 + threadIdx.x;
  float acc = 0.f;
  for (int k = 0; k < 640; ++k)
    acc += X[k * 640 + i] * X[k * 640 + j];
  if (i == j) acc += 0.001f;
  H[i * 640 + j] = acc;
}

// E = 0.5*(H11 + H33 + Y - Y^T)   (256x256)
__global__ void k_prepE(const float* __restrict__ H, const float* __restrict__ Y,
                        float* __restrict__ E) {
  int i = blockIdx.y * 16 + threadIdx.y;
  int j = blockIdx.x * 16 + threadIdx.x;
  E[i * 256 + j] = 0.5f * (H[i * 640 + j] + H[(384 + i) * 640 + 384 + j] +
                           Y[i * 256 + j] - Y[j * 256 + i]);
}

// Fm = H31 (256x256), B1 = H32 (256x128)
__global__ void k_prepFmB1(const float* __restrict__ H, float* __restrict__ Fm,
                           float* __restrict__ B1) {
  int j = blockIdx.y * 16 + threadIdx.y;   // < 256
  int c = blockIdx.x * 16 + threadIdx.x;   // < 384
  float v = H[(384 + j) * 640 + c];
  if (c < 256) Fm[j * 256 + c] = v;
  else         B1[j * 128 + (c - 256)] = v;
}

// C1t[k][i] = -H21[i][k]  (k<256, i<128), stored k-major for WMMA B loads
__global__ void k_prepC1t(const float* __restrict__ H, float* __restrict__ C1t) {
  int k = blockIdx.y * 16 + threadIdx.y;   // < 256
  int i = blockIdx.x * 16 + threadIdx.x;   // < 128
  C1t[k * 128 + i] = -H[(256 + i) * 640 + k];
}

// D11 = -tril(H22,-1), invLam[i] = 2 / H22[i][i]
__global__ void k_prepD11(const float* __restrict__ H, float* __restrict__ D11,
                          float* __restrict__ invLam) {
  int i = blockIdx.y * 16 + threadIdx.y;   // < 128
  int j = blockIdx.x * 16 + threadIdx.x;   // < 128
  float h = H[(256 + i) * 640 + 256 + j];
  D11[i * 128 + j] = (j < i) ? -h : 0.f;
  if (i == j) invLam[i] = 2.0f / h;
}

// dst[c*rows + r] = src[r*cols + c]
__global__ void k_transpose(const float* __restrict__ src, float* __restrict__ dst,
                            int rows, int cols) {
  int r = blockIdx.y * 16 + threadIdx.y;
  int c = blockIdx.x * 16 + threadIdx.x;
  if (r < rows && c < cols) dst[c * rows + r] = src[r * cols + c];
}

// Gauss-Jordan inversion with partial pivoting, one block of 256 threads.
__global__ void k_invert(const float* __restrict__ E, float* __restrict__ aug,
                         float* __restrict__ Einv) {
  int t = threadIdx.x;   // 0..255, owns row t
  for (int c = 0; c < 512; ++c)
    aug[t * 512 + c] = (c < 256) ? E[t * 256 + c] : ((c - 256) == t ? 1.f : 0.f);
  __syncthreads();
  __shared__ int piv;
  __shared__ float prow[512];
  for (int c = 0; c < 256; ++c) {
    if (t == 0) {
      int best = c; float bv = fabsf(aug[c * 512 + c]);
      for (int r = c + 1; r < 256; ++r) {
        float v = fabsf(aug[r * 512 + c]);
        if (v > bv) { bv = v; best = r; }
      }
      piv = best;
    }
    __syncthreads();
    int p = piv;
    if (p != c) {
      for (int cc = t; cc < 512; cc += 256) {
        float a = aug[c * 512 + cc], b = aug[p * 512 + cc];
        aug[c * 512 + cc] = b; aug[p * 512 + cc] = a;
      }
    }
    __syncthreads();
    float inv = 1.0f / aug[c * 512 + c];
    for (int cc = t; cc < 512; cc += 256) {
      float v = aug[c * 512 + cc] * inv;
      aug[c * 512 + cc] = v;
      prow[cc] = v;
    }
    __syncthreads();
    if (t != c) {
      float f = aug[t * 512 + c];
      if (f != 0.f)
        for (int cc = c; cc < 512; ++cc)
          aug[t * 512 + cc] -= f * prow[cc];
    }
    __syncthreads();
  }
  for (int cc = 0; cc < 256; ++cc)
    Einv[t * 256 + cc] = aug[t * 512 + 256 + cc];
}

// At[k*256+n] = sum_j M[j*ldm + k] * Einv[n*256 + j]  -> (E_inv @ M)^T, k-major
__global__ void k_einvmm(const float* __restrict__ M, int ldm,
                         const float* __restrict__ Einv, float* __restrict__ At,
                         int kcols) {
  int k = blockIdx.y * 16 + threadIdx.y;
  int n = blockIdx.x * 16 + threadIdx.x;
  if (k >= kcols) return;
  float acc = 0.f;
  for (int j = 0; j < 256; ++j)
    acc += M[j * ldm + k] * Einv[n * 256 + j];
  At[k * 256 + n] = acc;
}

// ---------------------------------------------------------------------------
// Persistent scan kernel: 32 workgroups x 128 threads (4 waves).
// Each workgroup owns batch rows [16*blockIdx.x, +16) for the full 511 steps.
// ---------------------------------------------------------------------------
__global__ __launch_bounds__(128) void ren_scan_kernel(
    const float* __restrict__ u_in,
    const float* __restrict__ Axt,  const float* __restrict__ Awt,
    const float* __restrict__ Aut,  const float* __restrict__ C2t,
    const float* __restrict__ D21t, const float* __restrict__ C1t,
    const float* __restrict__ D12t, const float* __restrict__ D11,
    const float* __restrict__ invLam,
    float* __restrict__ out) {
  __shared__ float xbuf0[16 * LDX];   // x slab (16 x 256), even steps read this
  __shared__ float xbuf1[16 * LDX];   // odd steps read this
  __shared__ float wsh[16 * LDW];     // w slab (16 x 128)
  __shared__ float bsh[16 * LDW];     // base slab (16 x 128)

  const int tid   = threadIdx.x;
  const int wave  = tid >> 5;
  const int lane  = tid & 31;
  const int mN    = lane & 15;   // m for A-frags, n for B/C-frags
  const int hf    = lane >> 4;   // K-pair / M-half selector
  const int mbase = blockIdx.x << 4;
  const int URS   = 512 * 64;    // stride between batch rows of u_in

  // ---- init: x0 = 0, y[:,0,:] = 0, base0 = u0 @ D12^T ----
  for (int idx = tid; idx < 16 * LDX; idx += 128) {
    xbuf0[idx] = 0.f; xbuf1[idx] = 0.f;
  }
  for (int idx = tid; idx < 16 * 64; idx += 128) {
    int m = idx >> 6, n = idx & 63;
    out[((size_t)(mbase + m) * 512 + 0) * 64 + n] = 0.f;
  }
  {
    const float* u0 = u_in + (size_t)mbase * URS;   // t = 0
    int nb = wave * 32;
    v8f acc[2] = {};
    wmma_multi<2>(u0, URS, D12t + nb, 128, 64, mN, hf, acc);
    for (int i = 0; i < 2; ++i)
      for (int r2 = 0; r2 < 8; ++r2)
        bsh[(r2 + 8 * hf) * LDW + nb + i * 16 + mN] = acc[i][r2];
  }
  __syncthreads();

  // One full time step: w-solve, x update, y + next base.
  auto step = [&](int t, const float* __restrict__ xcur,
                  float* __restrict__ xnxt) {
    // ---- stage 1: triangular tanh solve for w (8 lanes per batch row) ----
    {
      int m = tid >> 3, r = tid & 7;
      const float* brow = &bsh[m * LDW];
      float*       wrow = &wsh[m * LDW];
      for (int i = 0; i < 128; ++i) {
        const float* drow = D11 + i * 128;
        float a0 = 0.f, a1 = 0.f;
        int j = r;
        for (; j + 8 < i; j += 16) {           // dual chains: halve FMA depth
          a0 += wrow[j] * drow[j];
          a1 += wrow[j + 8] * drow[j + 8];
        }
        if (j < i) a0 += wrow[j] * drow[j];
        float acc = a0 + a1;
        acc += __shfl_xor(acc, 1);
        acc += __shfl_xor(acc, 2);
        acc += __shfl_xor(acc, 4);
        float val = tanhf((brow[i] + acc) * invLam[i]);
        if (r == 0) wrow[i] = val;
        asm volatile("" ::: "memory");  // keep LDS RAW ordered (in-wave DS ops in-order)
      }
    }
    __syncthreads();

    // ---- stage 2: x_new = x Ax^T + w Aw^T + u_t Au^T  (16 x 256) ----
    const float* ut = u_in + (size_t)mbase * URS + (size_t)t * 64;
    {
      int nb = wave * 64;                      // 4 tiles of 16 per wave
      v8f acc[4] = {};
      wmma_multi<4>(xcur, LDX, Axt + nb, 256, 256, mN, hf, acc);
      wmma_multi<4>(wsh,  LDW, Awt + nb, 256, 128, mN, hf, acc);
      wmma_multi<4>(ut,   URS, Aut + nb, 256,  64, mN, hf, acc);
      for (int i = 0; i < 4; ++i)
        for (int r2 = 0; r2 < 8; ++r2)
          xnxt[(r2 + 8 * hf) * LDX + nb + i * 16 + mN] = acc[i][r2];
    }
    __syncthreads();

    // ---- stage 3a: y_{t+1} = x_new C2^T + w D21^T  (16 x 64) ----
    {
      int nb = wave * 16;                      // 1 tile per wave
      v8f acc[1] = {};
      wmma_multi<1>(xnxt, LDX, C2t  + nb, 64, 256, mN, hf, acc);
      wmma_multi<1>(wsh,  LDW, D21t + nb, 64, 128, mN, hf, acc);
      for (int r2 = 0; r2 < 8; ++r2) {
        int mm = r2 + 8 * hf;
        out[((size_t)(mbase + mm) * 512 + (t + 1)) * 64 + nb + mN] = acc[0][r2];
      }
    }
    // ---- stage 3b: base_{t+1} = x_new C1^T + u_{t+1} D12^T  (16 x 128) ----
    {
      int nb = wave * 32;                      // 2 tiles per wave
      const float* ut1 = ut + 64;
      v8f acc[2] = {};
      wmma_multi<2>(xnxt, LDX, C1t  + nb, 128, 256, mN, hf, acc);
      wmma_multi<2>(ut1,  URS, D12t + nb, 128,  64, mN, hf, acc);
      for (int i = 0; i < 2; ++i)
        for (int r2 = 0; r2 < 8; ++r2)
          bsh[(r2 + 8 * hf) * LDW + nb + i * 16 + mN] = acc[i][r2];
    }
    __syncthreads();
  };

  // 511 steps = 255 double-steps + 1; buffers fixed at compile time.
  for (int s = 0; s < 255; ++s) {
    step(2 * s,     xbuf0, xbuf1);
    step(2 * s + 1, xbuf1, xbuf0);
  }
  step(510, xbuf0, xbuf1);
}

// ---------------------------------------------------------------------------
// Launch
// ---------------------------------------------------------------------------
extern "C" void kernel_launch(void* const* d_in, const int* in_sizes, int n_in,
                              void* d_out, int out_size, void* d_ws, size_t ws_size,
                              hipStream_t stream) {
  (void)in_sizes; (void)n_in; (void)out_size; (void)ws_size;
  const float* u_in = (const float*)d_in[0];
  const float* Xm   = (const float*)d_in[1];
  const float* Ym   = (const float*)d_in[2];
  const float* B2   = (const float*)d_in[3];
  const float* C2   = (const float*)d_in[4];
  const float* D21  = (const float*)d_in[5];
  const float* D12  = (const float*)d_in[6];
  float* out = (float*)d_out;
  float* wsp = (float*)d_ws;

  // workspace layout (floats), total ~3.9 MB
  float* H    = wsp + 0;        // 640*640
  float* E    = wsp + 409600;   // 256*256
  float* AUG  = wsp + 475136;   // 256*512
  float* EINV = wsp + 606208;   // 256*256
  float* FM   = wsp + 671744;   // 256*256
  float* B1M  = wsp + 737280;   // 256*128
  float* AXT  = wsp + 770048;   // 256*256
  float* AWT  = wsp + 835584;   // 128*256
  float* AUT  = wsp + 868352;   // 64*256
  float* C1T  = wsp + 884736;   // 256*128
  float* D12T = wsp + 917504;   // 64*128
  float* C2T  = wsp + 925696;   // 256*64
  float* D21T = wsp + 942080;   // 128*64
  float* D11M = wsp + 950272;   // 128*128
  float* ILAM = wsp + 966656;   // 128

  k_gram    <<<dim3(40, 40), dim3(16, 16), 0, stream>>>(Xm, H);
  k_prepE   <<<dim3(16, 16), dim3(16, 16), 0, stream>>>(H, Ym, E);
  k_prepFmB1<<<dim3(24, 16), dim3(16, 16), 0, stream>>>(H, FM, B1M);
  k_prepC1t <<<dim3(8, 16),  dim3(16, 16), 0, stream>>>(H, C1T);
  k_prepD11 <<<dim3(8, 8),   dim3(16, 16), 0, stream>>>(H, D11M, ILAM);
  k_transpose<<<dim3(4, 8),  dim3(16, 16), 0, stream>>>(D12, D12T, 128, 64);
  k_transpose<<<dim3(16, 4), dim3(16, 16), 0, stream>>>(C2,  C2T,  64, 256);
  k_transpose<<<dim3(8, 4),  dim3(16, 16), 0, stream>>>(D21, D21T, 64, 128);
  k_invert  <<<1, 256, 0, stream>>>(E, AUG, EINV);
  k_einvmm  <<<dim3(16, 16), dim3(16, 16), 0, stream>>>(FM,  256, EINV, AXT, 256);
  k_einvmm  <<<dim3(16, 8),  dim3(16, 16), 0, stream>>>(B1M, 128, EINV, AWT, 128);
  k_einvmm  <<<dim3(16, 4),  dim3(16, 16), 0, stream>>>(B2,   64, EINV, AUT,  64);

  ren_scan_kernel<<<32, 128, 0, stream>>>(u_in, AXT, AWT, AUT, C2T, D21T,
                                          C1T, D12T, D11M, ILAM, out);
}